// CompetitiveLayer_2_6279242186887
// MI455X (gfx1250) — compile-verified
//
#include <hip/hip_runtime.h>

typedef __attribute__((ext_vector_type(2))) float v2f;
typedef __attribute__((ext_vector_type(8))) float v8f;

#define WAVES          4
#define TPB            (32 * WAVES)
#define ROWS_PER_WAVE  16
#define ROWS_PER_BLOCK (ROWS_PER_WAVE * WAVES)   // 64
#define NITER          31                         // 30 no-grad solve iters + 1 differentiable iterate

// C = AF ⊗ K ⊗ BF after competitive-equilibrium fixed point.
// Per wave: 16 rows; matmuls done as 16x64 @ 64x64 via V_WMMA_F32_16X16X4_F32.
__global__ __launch_bounds__(TPB) void competitive_layer_kernel(
    const float* __restrict__ AT,     // (B, 64)
    const float* __restrict__ BT,     // (B, 64)
    const float* __restrict__ sqrtK,  // (64, 64)
    float* __restrict__ out)          // (B, 64, 64)
{
    // K row-major and K^T row-major: makes every 4x16 B-fragment a contiguous
    // float2 per lane (single ds_load_b64).
    __shared__ __align__(16) float Krm [64 * 64];
    __shared__ __align__(16) float KTrm[64 * 64];
    // Per-wave transpose scratch during iterations; final AF/BF staging at the end.
    __shared__ __align__(16) float AFs[ROWS_PER_BLOCK * 64];
    __shared__ __align__(16) float BFs[ROWS_PER_BLOCK * 64];

    const int tid  = threadIdx.x;
    const int lane = tid & 31;
    const int wave = tid >> 5;
    const int lm   = lane & 15;   // row within wave tile (A-layout), col within D-tile
    const int hi   = lane >> 4;   // lane half selects K-offset pair {0,1} vs {2,3}

    // ---- stage K = sqrt_K^2 (both orientations) ----
    for (int idx = tid; idx < 64 * 64; idx += TPB) {
        float s = sqrtK[idx];
        float k = s * s;
        Krm[idx] = k;
        KTrm[(idx & 63) * 64 + (idx >> 6)] = k;
    }
    __syncthreads();

    const int row0  = blockIdx.x * ROWS_PER_BLOCK + wave * ROWS_PER_WAVE;
    const int myrow = row0 + lm;

    // AT/BT/AF/BF held in registers in WMMA 16x4-f32 A-fragment layout:
    // reg[2*kc+c] = X[lm][4*kc + 2*hi + c]
    float at[32], bt[32], af[32], bf[32];
#pragma unroll
    for (int kc = 0; kc < 16; ++kc) {
        const float* pa = AT + (size_t)myrow * 64 + 4 * kc + 2 * hi;
        const float* pb = BT + (size_t)myrow * 64 + 4 * kc + 2 * hi;
        at[2 * kc] = pa[0]; at[2 * kc + 1] = pa[1];
        bt[2 * kc] = pb[0]; bt[2 * kc + 1] = pb[1];
        af[2 * kc] = at[2 * kc]; af[2 * kc + 1] = at[2 * kc + 1];  // carry init = (AT, BT)
        bf[2 * kc] = bt[2 * kc]; bf[2 * kc + 1] = bt[2 * kc + 1];
    }

    float* scrA = &AFs[wave * ROWS_PER_WAVE * 64];   // wave-private 16x64 scratch
    float* scrB = &BFs[wave * ROWS_PER_WAVE * 64];

    for (int it = 0; it < NITER; ++it) {
        // ---------- S = BF @ K^T  (B-fragments of K^T come from Krm) ----------
#pragma unroll
        for (int nt = 0; nt < 4; ++nt) {
            v8f acc = {0.f, 0.f, 0.f, 0.f, 0.f, 0.f, 0.f, 0.f};
            const float* kb = &Krm[(16 * nt + lm) * 64 + 2 * hi];
#pragma unroll
            for (int kc = 0; kc < 16; ++kc) {
                v2f a; a.x = bf[2 * kc]; a.y = bf[2 * kc + 1];
                v2f b; b.x = kb[4 * kc]; b.y = kb[4 * kc + 1];
                acc = __builtin_amdgcn_wmma_f32_16x16x4_f32(
                    false, a, false, b, (short)0, acc, false, false);
            }
#pragma unroll
            for (int v = 0; v < 8; ++v)   // D-layout: row = v+8*hi, col = 16*nt+lm
                scrA[(v + 8 * hi) * 64 + 16 * nt + lm] = acc[v];
        }
        __syncthreads();
        // transpose back to A-layout + elementwise AF = AT / (1 + S)
#pragma unroll
        for (int kc = 0; kc < 16; ++kc) {
            float s0 = scrA[lm * 64 + 4 * kc + 2 * hi];
            float s1 = scrA[lm * 64 + 4 * kc + 2 * hi + 1];
            af[2 * kc]     = at[2 * kc]     * __builtin_amdgcn_rcpf(1.0f + s0);
            af[2 * kc + 1] = at[2 * kc + 1] * __builtin_amdgcn_rcpf(1.0f + s1);
        }

        // ---------- S = AF @ K  (B-fragments of K come from KTrm) ----------
#pragma unroll
        for (int nt = 0; nt < 4; ++nt) {
            v8f acc = {0.f, 0.f, 0.f, 0.f, 0.f, 0.f, 0.f, 0.f};
            const float* kb = &KTrm[(16 * nt + lm) * 64 + 2 * hi];
#pragma unroll
            for (int kc = 0; kc < 16; ++kc) {
                v2f a; a.x = af[2 * kc]; a.y = af[2 * kc + 1];
                v2f b; b.x = kb[4 * kc]; b.y = kb[4 * kc + 1];
                acc = __builtin_amdgcn_wmma_f32_16x16x4_f32(
                    false, a, false, b, (short)0, acc, false, false);
            }
#pragma unroll
            for (int v = 0; v < 8; ++v)
                scrB[(v + 8 * hi) * 64 + 16 * nt + lm] = acc[v];
        }
        __syncthreads();
#pragma unroll
        for (int kc = 0; kc < 16; ++kc) {
            float s0 = scrB[lm * 64 + 4 * kc + 2 * hi];
            float s1 = scrB[lm * 64 + 4 * kc + 2 * hi + 1];
            bf[2 * kc]     = bt[2 * kc]     * __builtin_amdgcn_rcpf(1.0f + s0);
            bf[2 * kc + 1] = bt[2 * kc + 1] * __builtin_amdgcn_rcpf(1.0f + s1);
        }
    }

    // ---- stage final AF/BF row-major into LDS (overwrites scratch) ----
    __syncthreads();
#pragma unroll
    for (int kc = 0; kc < 16; ++kc) {
        scrA[lm * 64 + 4 * kc + 2 * hi]     = af[2 * kc];
        scrA[lm * 64 + 4 * kc + 2 * hi + 1] = af[2 * kc + 1];
        scrB[lm * 64 + 4 * kc + 2 * hi]     = bf[2 * kc];
        scrB[lm * 64 + 4 * kc + 2 * hi + 1] = bf[2 * kc + 1];
    }
    __syncthreads();

    // ---- C[r][i][j] = AF[r][i] * K[i][j] * BF[r][j]; contiguous float4 stores ----
    float4* out4 = reinterpret_cast<float4*>(out) +
                   (size_t)blockIdx.x * (ROWS_PER_BLOCK * 64 * 16);
    for (int idx = tid; idx < ROWS_PER_BLOCK * 64 * 16; idx += TPB) {
        int j4 = idx & 15;
        int i  = (idx >> 4) & 63;
        int r  = idx >> 10;
        float  a  = AFs[r * 64 + i];
        float4 kv = *reinterpret_cast<const float4*>(&Krm[i * 64 + j4 * 4]);
        float4 bv = *reinterpret_cast<const float4*>(&BFs[r * 64 + j4 * 4]);
        float4 o;
        o.x = a * kv.x * bv.x;
        o.y = a * kv.y * bv.y;
        o.z = a * kv.z * bv.z;
        o.w = a * kv.w * bv.w;
        out4[idx] = o;
    }
}

extern "C" void kernel_launch(void* const* d_in, const int* in_sizes, int n_in,
                              void* d_out, int out_size, void* d_ws, size_t ws_size,
                              hipStream_t stream) {
    const float* AT    = (const float*)d_in[0];
    const float* BT    = (const float*)d_in[1];
    const float* sqrtK = (const float*)d_in[2];
    float* out = (float*)d_out;

    const int Btot = in_sizes[0] / 64;           // 8192 rows
    const int grid = Btot / ROWS_PER_BLOCK;      // 128 blocks
    competitive_layer_kernel<<<grid, TPB, 0, stream>>>(AT, BT, sqrtK, out);
}